// MP_70600672411773
// MI455X (gfx1250) — compile-verified
//
#include <hip/hip_runtime.h>

// CDNA5 / gfx1250: wave32, WMMA 16x16x4 f32.
typedef __attribute__((ext_vector_type(2))) float v2f;
typedef __attribute__((ext_vector_type(8))) float v8f;

#define BB 64
#define TT 2048
#define DD 1024

// grid = (DD/128, BB), block = 256 threads (8 waves, each owns a 16-wide d tile)
__global__ __launch_bounds__(256) void MP_masked_mean_wmma(
    const float* __restrict__ feature,   // [B,T,D]
    const float* __restrict__ att_mask,  // [B,T]
    float* __restrict__ out)             // [B,D]
{
    const int b   = blockIdx.y;
    const int tid = threadIdx.x;

    // ---- 1) find length[b] = index of first negative mask + 1, or T ----
    __shared__ int s_min;
    if (tid == 0) s_min = 0x7FFFFFFF;
    __syncthreads();

    const float* mrow = att_mask + (size_t)b * TT;
    int lmin = 0x7FFFFFFF;
#pragma unroll
    for (int i = 0; i < TT / 256; ++i) {
        int t = tid + i * 256;              // coalesced strided scan
        float mv = mrow[t];
        if (mv < 0.0f && t < lmin) lmin = t;
    }
    if (lmin != 0x7FFFFFFF) atomicMin(&s_min, lmin);
    __syncthreads();
    const int first = s_min;
    const int len = (first == 0x7FFFFFFF) ? TT : (first + 1);  // uniform per block

    // ---- 2) masked time-sum via V_WMMA_F32_16X16X4_F32 ----
    // D[m,n] += sum_k A[m,k]*B[k,n];  A[m,k] = v[t+k] (same for all m),
    // B[k,n] = X[t+k, d0+n].  B VGPR layout (f32 4x16, 2 VGPRs):
    //   v.x : lanes 0-15 -> K=0, lanes 16-31 -> K=2
    //   v.y : lanes 0-15 -> K=1, lanes 16-31 -> K=3
    const int wave = tid >> 5;
    const int lane = tid & 31;
    const int half = lane >> 4;
    const int n    = lane & 15;
    const int d0   = blockIdx.x * 128 + wave * 16;

    const float* p = feature + (size_t)b * TT * DD + (size_t)d0 + n;
    const int r0 = 2 * half;       // K row feeding .x
    const int r1 = 2 * half + 1;   // K row feeding .y

    v8f acc0 = {}, acc1 = {}, acc2 = {}, acc3 = {};
    v2f ones; ones.x = 1.0f; ones.y = 1.0f;

    const int full = len & ~15;    // 4 chunks of K=4 per iteration
    for (int t = 0; t < full; t += 16) {
        v2f b0, b1, b2, b3;
        b0.x = __builtin_nontemporal_load(p + (size_t)(t +      r0) * DD);
        b0.y = __builtin_nontemporal_load(p + (size_t)(t +      r1) * DD);
        b1.x = __builtin_nontemporal_load(p + (size_t)(t + 4  + r0) * DD);
        b1.y = __builtin_nontemporal_load(p + (size_t)(t + 4  + r1) * DD);
        b2.x = __builtin_nontemporal_load(p + (size_t)(t + 8  + r0) * DD);
        b2.y = __builtin_nontemporal_load(p + (size_t)(t + 8  + r1) * DD);
        b3.x = __builtin_nontemporal_load(p + (size_t)(t + 12 + r0) * DD);
        b3.y = __builtin_nontemporal_load(p + (size_t)(t + 12 + r1) * DD);
        acc0 = __builtin_amdgcn_wmma_f32_16x16x4_f32(false, ones, false, b0,
                                                     (short)0, acc0, false, false);
        acc1 = __builtin_amdgcn_wmma_f32_16x16x4_f32(false, ones, false, b1,
                                                     (short)0, acc1, false, false);
        acc2 = __builtin_amdgcn_wmma_f32_16x16x4_f32(false, ones, false, b2,
                                                     (short)0, acc2, false, false);
        acc3 = __builtin_amdgcn_wmma_f32_16x16x4_f32(false, ones, false, b3,
                                                     (short)0, acc3, false, false);
    }
    // tail: chunks of 4 with 0/1 masks in A (rows stay < T, see analysis)
    for (int t = full; t < len; t += 4) {
        v2f a;
        a.x = ((t + r0) < len) ? 1.0f : 0.0f;
        a.y = ((t + r1) < len) ? 1.0f : 0.0f;
        v2f bm;
        bm.x = __builtin_nontemporal_load(p + (size_t)(t + r0) * DD);
        bm.y = __builtin_nontemporal_load(p + (size_t)(t + r1) * DD);
        acc0 = __builtin_amdgcn_wmma_f32_16x16x4_f32(false, a, false, bm,
                                                     (short)0, acc0, false, false);
    }

    // Every M-row of D is identical; acc[0] lanes 0-15 hold sum for d0+n.
    float s = (acc0[0] + acc1[0]) + (acc2[0] + acc3[0]);
    if (lane < 16) {
        out[(size_t)b * DD + d0 + n] = s / (float)len;
    }
}

extern "C" void kernel_launch(void* const* d_in, const int* in_sizes, int n_in,
                              void* d_out, int out_size, void* d_ws, size_t ws_size,
                              hipStream_t stream) {
    (void)in_sizes; (void)n_in; (void)out_size; (void)d_ws; (void)ws_size;
    const float* feature  = (const float*)d_in[0];
    const float* att_mask = (const float*)d_in[1];
    float* out = (float*)d_out;
    dim3 grid(DD / 128, BB);
    MP_masked_mean_wmma<<<grid, 256, 0, stream>>>(feature, att_mask, out);
}